// ReductionKernel_49082886258750
// MI455X (gfx1250) — compile-verified
//
#include <hip/hip_runtime.h>
#include <stdint.h>

// 2x2 stride-2 average pooling over fp32 NCHW (8,128,512,512) -> (8,128,256,256).
// Pure streaming kernel (~0.2 FLOP/byte): optimized for the MI455X 23.3 TB/s HBM
// path using gfx1250 async global->LDS copies (ASYNCcnt) with LDS double
// buffering, and non-temporal 64-bit stores for the write-once output.
//
// Tile T (T in [0, 131072)):
//   input  : x  + T*2048 floats  (4 contiguous input rows of 512  = 8 KB)
//   output : out + T*512 floats  (2 contiguous output rows of 256 = 2 KB)

typedef float f32x4 __attribute__((ext_vector_type(4)));
typedef float f32x2 __attribute__((ext_vector_type(2)));

#define TILE_IN_FLOATS   2048   // 4 rows * 512
#define TILE_OUT_FLOATS  512    // 2 rows * 256
#define THREADS          256    // 8 waves (wave32)
#define MAX_BLOCKS       8192

// Issue the per-thread portion (2 x b128 = 32 B) of one 8 KB tile copy into LDS.
// Uses the gfx1250 async copy: data goes straight to LDS, tracked by ASYNCcnt.
__device__ __forceinline__ void async_copy_tile(const float* __restrict__ src,
                                                uint32_t lds_byte_base, int tid) {
  uint64_t g0 = (uint64_t)(uintptr_t)src + (uint64_t)tid * 16u;  // bytes [16t, 16t+16)
  uint64_t g1 = g0 + 4096u;                                      // second half of tile
  uint32_t l0 = lds_byte_base + (uint32_t)tid * 16u;
  uint32_t l1 = l0 + 4096u;
  asm volatile("global_load_async_to_lds_b128 %0, %1, off"
               :: "v"(l0), "v"(g0) : "memory");
  asm volatile("global_load_async_to_lds_b128 %0, %1, off"
               :: "v"(l1), "v"(g1) : "memory");
}

__device__ __forceinline__ void compute_tile(const float* __restrict__ sm,
                                             float* __restrict__ dst, int tid) {
  // Thread t: output row r = t>>7 (0..1), output column pair j = t&127.
  // Needs input rows {2r, 2r+1} of the tile, input cols [4j, 4j+4).
  const int r = tid >> 7;
  const int j = tid & 127;
  const f32x4* row0 = (const f32x4*)(sm + (2 * r) * 512);
  const f32x4* row1 = (const f32x4*)(sm + (2 * r + 1) * 512);
  f32x4 a = row0[j];   // ds_load_b128, conflict-free (16 B/lane, contiguous)
  f32x4 b = row1[j];
  f32x2 o;
  o.x = 0.25f * ((a.x + a.y) + (b.x + b.y));
  o.y = 0.25f * ((a.z + a.w) + (b.z + b.w));
  // Write-once output: bypass caches with a non-temporal 64-bit store.
  __builtin_nontemporal_store(o, (f32x2*)(dst + r * 256 + 2 * j));
}

__global__ void __launch_bounds__(THREADS)
avgpool2x2_async_kernel(const float* __restrict__ x, float* __restrict__ out,
                        int num_tiles) {
  __shared__ __attribute__((aligned(16))) float smem[2][TILE_IN_FLOATS];
  const int tid = threadIdx.x;
  // addrspace(3) offset == low 32 bits of the generic pointer on gfx1250.
  const uint32_t lds0 = (uint32_t)(uintptr_t)&smem[0][0];
  const uint32_t lds1 = (uint32_t)(uintptr_t)&smem[1][0];

  int T = blockIdx.x;
  const int stride = gridDim.x;

  if (T < num_tiles) {
    async_copy_tile(x + (size_t)T * TILE_IN_FLOATS, lds0, tid);
  }

  int buf = 0;
  for (; T < num_tiles; T += stride, buf ^= 1) {
    const long Tn = (long)T + stride;
    const bool pf = (Tn < (long)num_tiles);
    if (pf) {
      // Prefetch next tile into the other buffer before consuming this one.
      async_copy_tile(x + (size_t)Tn * TILE_IN_FLOATS, buf ? lds0 : lds1, tid);
      // 2 copies for tile T complete (in-order), 2 for tile Tn may remain.
      asm volatile("s_wait_asynccnt 2" ::: "memory");
    } else {
      asm volatile("s_wait_asynccnt 0" ::: "memory");
    }
    __syncthreads();  // all waves' async data visible in LDS

    compute_tile(buf ? smem[1] : smem[0],
                 out + (size_t)T * TILE_OUT_FLOATS, tid);

    __syncthreads();  // done reading buf before it is overwritten next round
  }
}

extern "C" void kernel_launch(void* const* d_in, const int* in_sizes, int n_in,
                              void* d_out, int out_size, void* d_ws, size_t ws_size,
                              hipStream_t stream) {
  const float* x = (const float*)d_in[0];
  float* out = (float*)d_out;
  (void)in_sizes; (void)n_in; (void)d_ws; (void)ws_size;

  const int num_tiles = out_size / TILE_OUT_FLOATS;  // 131072 for this problem
  int grid = num_tiles < MAX_BLOCKS ? num_tiles : MAX_BLOCKS;
  if (grid < 1) grid = 1;

  avgpool2x2_async_kernel<<<dim3(grid), dim3(THREADS), 0, stream>>>(x, out, num_tiles);
}